// TritonGroupCrossAttention_6674379178647
// MI455X (gfx1250) — compile-verified
//
#include <hip/hip_runtime.h>
#include <hip/hip_bf16.h>

// ---- sizes (fixed by the reference) ----
// N=2 D=8 S=256 H=2048 NH=16 HD=128 K=8
// hs: (2,8,256,2048)  chunk_k/v: (14,16,8,256,128)  weights: (14,8)
// Wq/Wo: (2048,2048)  out: (2,2048,2048)

typedef __attribute__((ext_vector_type(2))) float v2f;
typedef __attribute__((ext_vector_type(8))) float v8f;

static __device__ __forceinline__ v8f wmma_f32(v2f a, v2f b, v8f c) {
  // D = A(16x4) * B(4x16) + C(16x16), fp32 everywhere.
  return __builtin_amdgcn_wmma_f32_16x16x4_f32(
      /*neg_a=*/false, a, /*neg_b=*/false, b,
      /*c_mod=*/(short)0, c, /*reuse_a=*/false, /*reuse_b=*/false);
}

// CDNA5 async copy: 16 bytes global -> LDS, tracked by ASYNCcnt.
// lds_off is the raw LDS byte offset (low 32 bits of a generic LDS pointer,
// which the LDS aperture uses as the address).
static __device__ __forceinline__ void async_copy16(unsigned lds_off,
                                                    const float* g) {
  unsigned long long ga = (unsigned long long)(size_t)g;
  asm volatile("global_load_async_to_lds_b128 %0, %1, off"
               :
               : "v"(lds_off), "v"(ga)
               : "memory");
}
static __device__ __forceinline__ void wait_async0() {
  asm volatile("s_wait_asynccnt 0x0" ::: "memory");
}

// row remap: logical row r in [0,3584) of the "chunks 1..7" space ->
// physical row in the full (n, D*S) = (n, 2048) space, skipping first chunk.
static __device__ __forceinline__ int remap_row(int r) {
  int n = r / 1792;
  return n * 2048 + 256 + (r - n * 1792);
}

// ---------------- RMSNorm ----------------
__global__ __launch_bounds__(256) void rmsnorm_kernel(
    const float* __restrict__ hs, const float* __restrict__ w,
    float* __restrict__ out) {
  const int row = blockIdx.x;
  const int tid = threadIdx.x;
  const float* x = hs + (size_t)row * 2048;
  float ss = 0.f;
  for (int i = tid; i < 2048; i += 256) { float v = x[i]; ss += v * v; }
  __shared__ float red[256];
  red[tid] = ss;
  __syncthreads();
  for (int off = 128; off > 0; off >>= 1) {
    if (tid < off) red[tid] += red[tid + off];
    __syncthreads();
  }
  const float inv = rsqrtf(red[0] * (1.0f / 2048.0f) + 1e-6f);
  float* o = out + (size_t)row * 2048;
  for (int i = tid; i < 2048; i += 256) o[i] = x[i] * inv * w[i];
}

// -------- TN GEMM: C = A * B^T, async double-buffered LDS tiles ------------
// A: Mx2048 (rows optionally remapped), B: 2048x2048 (row-major, K-contig),
// C: Mx2048 (rows optionally remapped). Block tile 64(M) x 128(N), BK=32.
// 8 waves; each wave computes a 32x32 block = 4 WMMA accumulators.
template <int REMAP_A, int REMAP_OUT>
__global__ __launch_bounds__(256) void gemm_async(const float* __restrict__ A,
                                                  const float* __restrict__ B,
                                                  float* __restrict__ C) {
  constexpr int K = 2048;
  constexpr int PA = 36;                       // padded LDS row pitch (floats)
  constexpr int BUF = 64 * PA + 128 * PA;      // floats per buffer (A then B)
  __shared__ float smem[2 * BUF];

  const int bn = blockIdx.x * 128;
  const int bm = blockIdx.y * 64;
  const int tid = threadIdx.x;
  const int lane = tid & 31;
  const int wave = tid >> 5;
  const int lm = lane & 15;            // M (A) / N (B) index within subtile
  const int kh = (lane >> 4) * 2;      // K half-select per ISA A/B layout
  const int m0 = (wave & 1) * 2;       // first of 2 M-subtiles (of 4)
  const int n0 = (wave >> 1) * 2;      // first of 2 N-subtiles (of 8)

  // cooperative async-load assignments
  const int ar = tid >> 2;             // A tile row 0..63
  const int ac = (tid & 3) * 4;        // A col group (2 b128 per thread)
  const int br = tid >> 1;             // B tile row 0..127
  const int bc = (tid & 1) * 16;       // B col half (4 b128 per thread)
  int arowg = bm + ar;
  if (REMAP_A) arowg = remap_row(arowg);
  const float* Ap = A + (size_t)arowg * K + ac;
  const float* Bp = B + (size_t)(bn + br) * K + bc;

  const unsigned lbase = (unsigned)(size_t)smem;  // LDS byte offset of smem

  auto issue = [&](int buf, int k0) {
    const unsigned la = lbase + (unsigned)(buf * BUF) * 4u;
    const unsigned lb = la + 64u * PA * 4u;
    async_copy16(la + (unsigned)(ar * PA + ac) * 4u, Ap + k0);
    async_copy16(la + (unsigned)(ar * PA + ac + 16) * 4u, Ap + k0 + 16);
#pragma unroll
    for (int j = 0; j < 4; j++)
      async_copy16(lb + (unsigned)(br * PA + bc + 4 * j) * 4u,
                   Bp + k0 + 4 * j);
  };

  v8f c[2][2] = {{{}, {}}, {{}, {}}};

  issue(0, 0);
  int buf = 0;
  for (int k0 = 0; k0 < K; k0 += 32) {
    wait_async0();       // this iteration's tile has landed in LDS
    __syncthreads();     // ...for every wave; also fences last iter's compute
    if (k0 + 32 < K) issue(buf ^ 1, k0 + 32);  // overlap next tile with math
    const float* Sa = smem + buf * BUF;
    const float* Sb = Sa + 64 * PA;
#pragma unroll
    for (int kk = 0; kk < 32; kk += 4) {
      v2f a[2], b[2];
#pragma unroll
      for (int i = 0; i < 2; i++) {
        float2 t2 = *(const float2*)&Sa[((m0 + i) * 16 + lm) * PA + kk + kh];
        a[i].x = t2.x; a[i].y = t2.y;
      }
#pragma unroll
      for (int j = 0; j < 2; j++) {
        float2 t2 = *(const float2*)&Sb[((n0 + j) * 16 + lm) * PA + kk + kh];
        b[j].x = t2.x; b[j].y = t2.y;
      }
#pragma unroll
      for (int i = 0; i < 2; i++)
#pragma unroll
        for (int j = 0; j < 2; j++) c[i][j] = wmma_f32(a[i], b[j], c[i][j]);
    }
    buf ^= 1;
  }

#pragma unroll
  for (int i = 0; i < 2; i++)
#pragma unroll
    for (int j = 0; j < 2; j++)
#pragma unroll
      for (int v = 0; v < 8; v++) {
        int r = bm + (m0 + i) * 16 + v + 8 * (lane >> 4);
        int outr = REMAP_OUT ? remap_row(r) : r;
        C[(size_t)outr * 2048 + bn + (n0 + j) * 16 + lm] = c[i][j][v];
      }
}

// ---------------- Fused attention ----------------
// One block per (b, h, 64-query tile). Q tile in LDS; loop 8 chunks:
// scores = q*K^T/sqrt(128) -> LDS, softmax-one, out += w[b,k] * (P @ V).
__global__ __launch_bounds__(256) void attn_kernel(
    const float* __restrict__ q, const float* __restrict__ ck,
    const float* __restrict__ cvp, const float* __restrict__ wts,
    float* __restrict__ o) {
  extern __shared__ float smem[];
  float* qs   = smem;               // 64 x 132 (padded)
  float* ps   = smem + 64 * 132;    // 64 x 260 (padded)
  float* red1 = ps + 64 * 260;      // 256
  float* red2 = red1 + 256;         // 256

  const int blk = blockIdx.x;       // (b*16 + h)*4 + stile
  const int stile = blk & 3;
  const int h = (blk >> 2) & 15;
  const int b = blk >> 6;
  const int tid = threadIdx.x;
  const int lane = tid & 31;
  const int wave = tid >> 5;
  const int lm = lane & 15;
  const int kh = (lane >> 4) * 2;
  const int msub = wave & 3;        // 4 M-subtiles of the 64-query tile
  const int nhalf = wave >> 2;      // splits the N dimension between wave pairs

  // stage Q tile (64 x 128) into LDS
  {
    const size_t qbase = (size_t)(b * 256 + stile * 64) * 2048 + h * 128;
    int r = tid >> 2;
    int c = (tid & 3) * 32;
    for (int i = 0; i < 32; i += 4) {
      float4 v = *(const float4*)(q + qbase + (size_t)r * 2048 + c + i);
      qs[r * 132 + c + i + 0] = v.x;
      qs[r * 132 + c + i + 1] = v.y;
      qs[r * 132 + c + i + 2] = v.z;
      qs[r * 132 + c + i + 3] = v.w;
    }
  }
  __syncthreads();

  v8f oacc[4] = {{}, {}, {}, {}};
  const size_t kvbase0 = (size_t)(b * 16 + h) * 8 * (256 * 128);

  for (int kc = 0; kc < 8; kc++) {
    const float* Kp = ck + kvbase0 + (size_t)kc * (256 * 128);
    const float* Vp = cvp + kvbase0 + (size_t)kc * (256 * 128);

    // ---- scores: each wave does 16(M) x 128(N) of the 64x256 score tile ----
    v8f cs[8] = {{}, {}, {}, {}, {}, {}, {}, {}};
    for (int d = 0; d < 128; d += 4) {
      v2f a;
      {
        float2 t2 = *(const float2*)&qs[(msub * 16 + lm) * 132 + d + kh];
        a.x = t2.x; a.y = t2.y;
      }
#pragma unroll
      for (int j = 0; j < 8; j++) {
        int n = nhalf * 128 + j * 16 + lm;   // kv position
        float2 t2 = *(const float2*)(Kp + (size_t)n * 128 + d + kh);
        v2f bb; bb.x = t2.x; bb.y = t2.y;
        cs[j] = wmma_f32(a, bb, cs[j]);
      }
    }
    const float scale = 0.08838834764831845f;  // 1/sqrt(128)
#pragma unroll
    for (int j = 0; j < 8; j++)
#pragma unroll
      for (int v = 0; v < 8; v++) {
        int m = msub * 16 + v + 8 * (lane >> 4);
        int n = nhalf * 128 + j * 16 + lm;
        ps[m * 260 + n] = cs[j][v] * scale;
      }
    __syncthreads();

    // ---- softmax-one over the 256 kv positions of this chunk ----
    {
      int r = tid >> 2;
      int c0 = (tid & 3) * 64;
      float mx = -3.4e38f;
      for (int c = 0; c < 64; c++) mx = fmaxf(mx, ps[r * 260 + c0 + c]);
      red1[tid] = mx;
      __syncthreads();
      float rm = fmaxf(fmaxf(red1[r * 4 + 0], red1[r * 4 + 1]),
                       fmaxf(red1[r * 4 + 2], red1[r * 4 + 3]));
      float sum = 0.f;
      for (int c = 0; c < 64; c++) {
        float e = __expf(ps[r * 260 + c0 + c] - rm);
        ps[r * 260 + c0 + c] = e;
        sum += e;
      }
      red2[tid] = sum;
      __syncthreads();
      float denom = red2[r * 4 + 0] + red2[r * 4 + 1] + red2[r * 4 + 2] +
                    red2[r * 4 + 3] + __expf(-rm);   // softmax-one extra term
      float inv = 1.0f / denom;
      for (int c = 0; c < 64; c++) ps[r * 260 + c0 + c] *= inv;
    }
    __syncthreads();

    // ---- P @ V: each wave does 16(M) x 64(HD) of the 64x128 output ----
    v8f co[4] = {{}, {}, {}, {}};
    for (int t = 0; t < 256; t += 4) {
      v2f a;
      {
        float2 t2 = *(const float2*)&ps[(msub * 16 + lm) * 260 + t + kh];
        a.x = t2.x; a.y = t2.y;
      }
#pragma unroll
      for (int j = 0; j < 4; j++) {
        int n = (nhalf * 4 + j) * 16 + lm;   // head-dim index
        v2f bb;
        bb.x = Vp[(size_t)(t + kh) * 128 + n];
        bb.y = Vp[(size_t)(t + kh + 1) * 128 + n];
        co[j] = wmma_f32(a, bb, co[j]);
      }
    }
    const float wk = wts[b * 8 + kc];
#pragma unroll
    for (int j = 0; j < 4; j++)
#pragma unroll
      for (int v = 0; v < 8; v++) oacc[j][v] += wk * co[j][v];
    __syncthreads();  // protect ps before next chunk overwrites it
  }

  // write attention output: row = b*256 + s, col = h*128 + dd
#pragma unroll
  for (int j = 0; j < 4; j++)
#pragma unroll
    for (int v = 0; v < 8; v++) {
      int m = msub * 16 + v + 8 * (lane >> 4);
      int n = (nhalf * 4 + j) * 16 + lm;
      o[(size_t)(b * 256 + stile * 64 + m) * 2048 + h * 128 + n] = oacc[j][v];
    }
}

// ---------------- zero the first-chunk output rows ----------------
__global__ __launch_bounds__(256) void zero_first_chunk(float* __restrict__ out) {
  int i = blockIdx.x * 256 + threadIdx.x;   // 2 * 256 * 2048 = 2^20 elements
  int n = i >> 19;                          // 256*2048 = 2^19 per batch
  int rem = i & ((1 << 19) - 1);
  out[(size_t)n * (2048 * 2048) + rem] = 0.0f;
}

extern "C" void kernel_launch(void* const* d_in, const int* in_sizes, int n_in,
                              void* d_out, int out_size, void* d_ws,
                              size_t ws_size, hipStream_t stream) {
  const float* hs = (const float*)d_in[0];   // (2,8,256,2048)
  const float* ck = (const float*)d_in[1];   // (14,16,8,256,128)
  const float* cv = (const float*)d_in[2];   // (14,16,8,256,128)
  const float* wt = (const float*)d_in[3];   // (14,8)
  const float* rw = (const float*)d_in[4];   // (2048,)
  const float* Wq = (const float*)d_in[5];   // (2048,2048)
  const float* Wo = (const float*)d_in[6];   // (2048,2048)
  float* out = (float*)d_out;                // (2,2048,2048)

  float* hs_norm = (float*)d_ws;                // 4096*2048
  float* q_ws = hs_norm + (size_t)4096 * 2048;  // 3584*2048
  float* o_ws = q_ws + (size_t)3584 * 2048;     // 3584*2048

  // 1) RMSNorm all 4096 rows
  rmsnorm_kernel<<<4096, 256, 0, stream>>>(hs, rw, hs_norm);

  // 2) q = hs_norm @ Wq^T, computed only for chunks 1..7 (A-row remap)
  gemm_async<1, 0><<<dim3(16, 56), 256, 0, stream>>>(hs_norm, Wq, q_ws);

  // 3) fused attention over 8 retrieved chunks per (b,h)
  attn_kernel<<<14 * 16 * 4, 256, (64 * 132 + 64 * 260 + 512) * sizeof(float),
                stream>>>(q_ws, ck, cv, wt, o_ws);

  // 4) zero first-chunk output rows
  zero_first_chunk<<<4096, 256, 0, stream>>>(out);

  // 5) final = o_ws @ Wo^T scattered into out[:, S:, :] (out-row remap)
  gemm_async<0, 1><<<dim3(16, 56), 256, 0, stream>>>(o_ws, Wo, out);
}